// Model_39676907884794
// MI455X (gfx1250) — compile-verified
//
#include <hip/hip_runtime.h>

typedef __attribute__((ext_vector_type(2))) float v2f;
typedef __attribute__((ext_vector_type(4))) float v4f;
typedef __attribute__((ext_vector_type(8))) float v8f;
typedef __attribute__((ext_vector_type(2))) unsigned v2u;

#define HW   (512u * 512u)   // 262144 pixels per channel plane
#define BSTR (3u * HW)       // batch stride of x (NCHW, C=3)
#define NPIX (16u * HW)      // 4,194,304 total pixels

// 1 = use v_permlane16_swap_b32 (CDNA5 lane-half swap, pure VALU)
// 0 = fallback: ds_swizzle SWAPX16 + cndmask role selection
#define USE_PERMLANE_SWAP 1

struct PrepPtrs {
  const float* w[7];  // w2..w8 (each 8x8 row-major)
  const float* b[7];  // b2..b8 (each 8)
};

// Compose the linear chain w8∘...∘w2 into M (8x8) + C (8), then emit
// Wext[16][4]: rows 0-7 = [w1 | b1], rows 8-15 = [M·w1 | M·b1 + C].
__global__ void prep_kernel(PrepPtrs p, const float* __restrict__ w1,
                            const float* __restrict__ b1,
                            float* __restrict__ wext) {
  __shared__ float M[64], C[8], T[64], TC[8];
  const int tid = threadIdx.x;            // 64 threads
  const int i = tid >> 3, j = tid & 7;
  M[tid] = p.w[0][tid];
  if (j == 0) C[i] = p.b[0][i];
  __syncthreads();
  for (int k = 1; k < 7; ++k) {
    float s = 0.f;
    for (int t = 0; t < 8; ++t) s += p.w[k][i * 8 + t] * M[t * 8 + j];
    T[tid] = s;
    if (j == 0) {
      float sc = p.b[k][i];
      for (int t = 0; t < 8; ++t) sc += p.w[k][i * 8 + t] * C[t];
      TC[i] = sc;
    }
    __syncthreads();
    M[tid] = T[tid];
    if (j == 0) C[i] = TC[i];
    __syncthreads();
  }
  const int row = tid >> 2, col = tid & 3;
  float val;
  if (row < 8) {
    val = (col < 3) ? w1[row * 3 + col] : b1[row];
  } else {
    const int ii = row - 8;
    if (col < 3) {
      float s = 0.f;
      for (int t = 0; t < 8; ++t) s += M[ii * 8 + t] * w1[t * 3 + col];
      val = s;
    } else {
      float s = C[ii];
      for (int t = 0; t < 8; ++t) s += M[ii * 8 + t] * b1[t];
      val = s;
    }
  }
  wext[tid] = val;
}

__device__ __forceinline__ float fsig(float x) {
  // sigmoid(x) = 1 / (1 + exp(-x)) via native v_exp_f32 / v_rcp_f32
  float e = __builtin_amdgcn_exp2f(-1.4426950408889634f * x);
  return __builtin_amdgcn_rcpf(1.0f + e);
}

// One wave handles 2 x 128 pixels; each 128-px chunk = 8 WMMA tiles of 16 px.
// WMMA: D(16 out-ch x 16 px) = Wext(16x4) x [R;G;B;1](4x16), bias folded in K=3.
__global__ void __launch_bounds__(256) fused_kernel(
    const float* __restrict__ x, const float* __restrict__ wext,
    const float* __restrict__ wres, const float* __restrict__ bres,
    float* __restrict__ out) {
  const int lane = threadIdx.x & 31;
  const int ll = lane & 15;
  const int hi = lane >> 4;                      // 0 = lanes 0-15, 1 = lanes 16-31
  const unsigned waveId = blockIdx.x * 8u + (threadIdx.x >> 5);
  const unsigned P0 = waveId * 256u;             // 2 chunks of 128 px

  // A operand: 16x4 f32 -> lane l<16: K=0,1 ; lane l+16: K=2,3 (row M = l%16)
  v2f A;
  A.x = wext[ll * 4 + (hi ? 2 : 0)];
  A.y = wext[ll * 4 + (hi ? 3 : 1)];

  float wr[8];
#pragma unroll
  for (int r = 0; r < 8; ++r) wr[r] = wres[r];
  const float br = bres[0];

#pragma unroll
  for (int it = 0; it < 2; ++it) {
    const unsigned P = P0 + 128u * it;           // 128 | HW -> chunk in one image
    const unsigned b = P / HW;
    const unsigned q = P - b * HW;
    const float* plane0 = x + (size_t)b * BSTR + (hi ? 2u * HW : 0u) + q; // R / Blue
    const float* planeG = x + (size_t)b * BSTR + HW + q;                  // G

    // Coalesced B128 loads: component t of lane ll = pixel P + 64*s + 4*ll + t
    v4f p0[2], p1[2];
#pragma unroll
    for (int s = 0; s < 2; ++s) {
      p0[s] = *(const v4f*)(plane0 + 64 * s + 4 * ll);
      v4f g = *(const v4f*)(planeG + 64 * s + 4 * ll);
#pragma unroll
      for (int t = 0; t < 4; ++t) p1[s][t] = hi ? 1.0f : g[t]; // K=3 row = 1.0
    }

    v8f D[2][4];
#pragma unroll
    for (int s = 0; s < 2; ++s)
#pragma unroll
      for (int t = 0; t < 4; ++t) {
        v2f B;
        B.x = p0[s][t];   // K=0 (R) on low lanes / K=2 (Blue) on high lanes
        B.y = p1[s][t];   // K=1 (G) on low lanes / K=3 (1.0) on high lanes
        v8f c = {};
        D[s][t] = __builtin_amdgcn_wmma_f32_16x16x4_f32(
            false, A, false, B, (short)0, c, false, false);
      }

    // Epilogue: low lanes finish tiles s=0, high lanes tiles s=1.
    v4f ov;
#pragma unroll
    for (int t = 0; t < 4; ++t) {
      float acc = br;
#pragma unroll
      for (int r = 0; r < 8; ++r) {
#if USE_PERMLANE_SWAP
        // dst[16..31] <-> src[0..15]:
        //   res[0] = v1 of my assigned tile in every lane
        //   res[1] = v8 of my assigned tile in every lane
        v2u res = __builtin_amdgcn_permlane16_swap(
            __float_as_uint(D[0][t][r]), __float_as_uint(D[1][t][r]),
            false, false);
        float v1v = __uint_as_float(res[0]);
        float v8v = __uint_as_float(res[1]);
#else
        // ds_swizzle SWAPX16 (xor 0x10): offer what the partner needs
        float give = hi ? D[0][t][r] : D[1][t][r];
        float got = __uint_as_float((unsigned)__builtin_amdgcn_ds_swizzle(
            (int)__float_as_uint(give), 0x401F));
        float v1v = hi ? got : D[0][t][r];
        float v8v = hi ? D[1][t][r] : got;
#endif
        acc += wr[r] * (v1v * fsig(v8v));
      }
      ov[t] = acc * fsig(acc);               // silu(v11)
    }
    // Lane's 4 outputs are contiguous -> full-wave 512B coalesced store
    *(v4f*)(out + P + 64u * hi + 4u * ll) = ov;
  }
}

extern "C" void kernel_launch(void* const* d_in, const int* in_sizes, int n_in,
                              void* d_out, int out_size, void* d_ws, size_t ws_size,
                              hipStream_t stream) {
  (void)in_sizes; (void)n_in; (void)out_size; (void)ws_size;
  const float* x  = (const float*)d_in[0];
  const float* w1 = (const float*)d_in[1];
  const float* b1 = (const float*)d_in[2];
  PrepPtrs p;
  for (int k = 0; k < 7; ++k) {
    p.w[k] = (const float*)d_in[3 + 2 * k];
    p.b[k] = (const float*)d_in[4 + 2 * k];
  }
  const float* wres = (const float*)d_in[17];
  const float* bres = (const float*)d_in[18];
  float* wext = (float*)d_ws;

  prep_kernel<<<1, 64, 0, stream>>>(p, w1, b1, wext);
  fused_kernel<<<NPIX / 2048u, 256, 0, stream>>>(x, wext, wres, bres,
                                                 (float*)d_out);
}